// LookModule_30150670418654
// MI455X (gfx1250) — compile-verified
//
#include <hip/hip_runtime.h>
#include <hip/hip_bf16.h>
#include <cstddef>

// ---------------------------------------------------------------------------
// Problem constants (from reference)
// ---------------------------------------------------------------------------
#define BSZ   8
#define TT    5
#define EE    128
#define NCAM  4
#define NZ    15
#define NP    (TT + 4)          // 9 look points
#define NQ    (NP * NZ)         // 135 queries
#define BN    (BSZ * NCAM)      // 32
#define DD    256
#define HEADS 8
#define HD    32
#define LVLS  4
#define PTS   4
#define QDIM  1799
#define BASED 775               // 4 + 3 + 128 + 128 + 512
#define STOT  4760

typedef __attribute__((ext_vector_type(2))) float v2f;
typedef __attribute__((ext_vector_type(8))) float v8f;

// ---------------------------------------------------------------------------
// Kernel A: build base query columns + camera projection (rpc, mask)
// one block per (b, q); 128 threads
// ---------------------------------------------------------------------------
__global__ void build_query_proj(const float* __restrict__ wp,
                                 const float* __restrict__ ctrl,
                                 const float* __restrict__ meas,
                                 const float* __restrict__ flat,
                                 const float* __restrict__ l2i,
                                 const float* __restrict__ ida,
                                 const float* __restrict__ temb,
                                 const float* __restrict__ semb,
                                 const int*   __restrict__ ih,
                                 const int*   __restrict__ iw,
                                 float* __restrict__ qfull,
                                 float* __restrict__ rpc,
                                 float* __restrict__ maskf) {
    const int bq = blockIdx.x;            // 0 .. BSZ*NQ-1
    const int b = bq / NQ, q = bq % NQ;
    const int p = q / NZ, zi = q % NZ;

    const float sp[4][2] = {{5.f, 0.f}, {0.f, -5.f}, {0.f, 5.f}, {-5.f, 0.f}};
    float x, y;
    if (p < TT) { x = wp[(b * TT + p) * 2 + 0]; y = wp[(b * TT + p) * 2 + 1]; }
    else        { x = sp[p - TT][0];            y = sp[p - TT][1]; }
    const float z = -4.0f + (float)zi;    // linspace(-4, 10, 15), step 1

    const int tid = threadIdx.x;
    if (tid < NCAM) {
        const int c = tid;
        const float* L = l2i + (size_t)(b * NCAM + c) * 16;
        float pc[4];
#pragma unroll
        for (int i = 0; i < 4; ++i)
            pc[i] = L[i*4+0]*x + L[i*4+1]*y + L[i*4+2]*z + L[i*4+3];
        const float zc = fmaxf(pc[2], 1e-5f);
        float p2[4] = {pc[0]/zc, pc[1]/zc, pc[2], pc[3]};
        const float* I = ida + (size_t)(b * NCAM + c) * 16;
        float p3[3];
#pragma unroll
        for (int i = 0; i < 3; ++i)
            p3[i] = I[i*4+0]*p2[0] + I[i*4+1]*p2[1] + I[i*4+2]*p2[2] + I[i*4+3]*p2[3];
        const float rx = p3[0] / (float)iw[0];
        const float ry = p3[1] / (float)ih[0];
        const int idx = (b * NCAM + c) * NQ + q;
        rpc[idx*2+0] = rx;
        rpc[idx*2+1] = ry;
        maskf[idx] = (p3[2] > 1e-5f && ry > 0.f && ry < 1.f && rx > 0.f && rx < 1.f)
                         ? 1.f : 0.f;
    }

    for (int col = tid; col < BASED; col += blockDim.x) {
        float v;
        if (col < 4)        v = (p < TT) ? ctrl[(b * TT + p) * 4 + col] : 0.f;
        else if (col == 4)  v = x;
        else if (col == 5)  v = y;
        else if (col == 6)  v = z;
        else if (col < 7 + EE) {
            const int e = col - 7;
            v = (p < TT) ? temb[p * EE + e] : semb[(p - TT) * EE + e];
        }
        else if (col < 7 + EE + 128) v = meas[b * 128 + (col - 7 - EE)];
        else                         v = flat[b * 512 + (col - 7 - EE - 128)];
#pragma unroll
        for (int c = 0; c < NCAM; ++c)
            qfull[(size_t)((b * NCAM + c) * NQ + q) * QDIM + col] = v;
    }
}

// ---------------------------------------------------------------------------
// Kernel B: bilinear sample 4 FPN levels into qfull[:, 775:1799]  (ch*4 + l)
// ---------------------------------------------------------------------------
__global__ void sample_img_feats(const float* __restrict__ f0,
                                 const float* __restrict__ f1,
                                 const float* __restrict__ f2,
                                 const float* __restrict__ f3,
                                 const float* __restrict__ rpc,
                                 float* __restrict__ qfull) {
    const int idx = blockIdx.x * blockDim.x + threadIdx.x;
    const int total = BN * NQ * LVLS * DD;
    if (idx >= total) return;
    const int ch = idx & 255;
    const int l  = (idx >> 8) & 3;
    const int t  = idx >> 10;               // bn*NQ + q
    const int bn = t / NQ;

    const float rx = rpc[t * 2 + 0];
    const float ry = rpc[t * 2 + 1];

    const float* f; int H, W;
    switch (l) {
        case 0:  f = f0; H = 32; W = 112; break;
        case 1:  f = f1; H = 16; W = 56;  break;
        case 2:  f = f2; H = 8;  W = 28;  break;
        default: f = f3; H = 4;  W = 14;  break;
    }
    const float x = rx * (float)W - 0.5f;
    const float y = ry * (float)H - 0.5f;
    const int x0 = (int)floorf(x), y0 = (int)floorf(y);
    const float wx = x - (float)x0, wy = y - (float)y0;
    const float* base = f + ((size_t)bn * DD + ch) * (size_t)(H * W);

    auto g = [&](int xi, int yi) -> float {
        return (xi >= 0 && xi < W && yi >= 0 && yi < H) ? base[yi * W + xi] : 0.f;
    };
    const float v = g(x0,   y0  ) * (1.f - wx) * (1.f - wy)
                  + g(x0+1, y0  ) * wx         * (1.f - wy)
                  + g(x0,   y0+1) * (1.f - wx) * wy
                  + g(x0+1, y0+1) * wx         * wy;

    qfull[(size_t)t * QDIM + BASED + ch * 4 + l] = v;
}

// ---------------------------------------------------------------------------
// WMMA f32 GEMM:  C[M,N] = A[M,K] @ B[K,N] + bias[N]
// row-major; M % 16 == 0, N % 64 == 0; K arbitrary (peeled tail).
// 128 threads = 4 waves; each wave owns a 16x64 strip (4 accumulators).
// Main K-loop has NO guards -> no exec divergence, loads batch before WMMA.
// A(16x4 f32) lane map: lane = (k>>1)*16 + m, vgpr = k&1
// B(4x16)     lane map: lane = (k>>1)*16 + n, vgpr = k&1
// C/D(16x16)  vgpr r : M = r + 8*(lane>=16), N = lane&15
// ---------------------------------------------------------------------------
__global__ void gemm_wmma_f32(const float* __restrict__ A,
                              const float* __restrict__ B,
                              const float* __restrict__ bias,
                              float* __restrict__ C,
                              int M, int N, int K) {
    const int lane  = threadIdx.x & 31;
    const int wave  = threadIdx.x >> 5;
    const int mtile = blockIdx.x * 4 + wave;       // wave-uniform
    if (mtile * 16 >= M) return;                   // whole wave exits together

    const int m0 = mtile * 16;
    const int n0 = blockIdx.y * 64;                // 4 N-tiles of 16
    const int mi = lane & 15;                      // M (for A) / N (for B) index
    const int kh = (lane >> 4) << 1;               // 0 or 2: K sub-offset

    v8f acc0 = {0.f,0.f,0.f,0.f,0.f,0.f,0.f,0.f};
    v8f acc1 = acc0, acc2 = acc0, acc3 = acc0;

    const float* Arow = A + (size_t)(m0 + mi) * K;
    const float* B0   = B + n0 + mi;
    const int K4 = K & ~3;

#pragma unroll 2
    for (int k0 = 0; k0 < K4; k0 += 4) {
        const int ka = k0 + kh;
        const float* Br0 = B0 + (size_t)ka * N;
        const float* Br1 = Br0 + N;
        v2f a, b0, b1, b2, b3;
        a[0]  = Arow[ka];   a[1]  = Arow[ka + 1];
        b0[0] = Br0[0];     b0[1] = Br1[0];
        b1[0] = Br0[16];    b1[1] = Br1[16];
        b2[0] = Br0[32];    b2[1] = Br1[32];
        b3[0] = Br0[48];    b3[1] = Br1[48];
        acc0 = __builtin_amdgcn_wmma_f32_16x16x4_f32(false, a, false, b0, (short)0, acc0, false, false);
        acc1 = __builtin_amdgcn_wmma_f32_16x16x4_f32(false, a, false, b1, (short)0, acc1, false, false);
        acc2 = __builtin_amdgcn_wmma_f32_16x16x4_f32(false, a, false, b2, (short)0, acc2, false, false);
        acc3 = __builtin_amdgcn_wmma_f32_16x16x4_f32(false, a, false, b3, (short)0, acc3, false, false);
    }

    if (K4 < K) {                                  // single guarded tail (K%4 != 0)
        const int ka = K4 + kh;
        const bool ok0 = (ka     < K);
        const bool ok1 = (ka + 1 < K);
        const float* Br0 = B0 + (size_t)ka * N;
        const float* Br1 = Br0 + N;
        v2f a, b0, b1, b2, b3;
        a[0]  = ok0 ? Arow[ka]     : 0.f;
        a[1]  = ok1 ? Arow[ka + 1] : 0.f;
        b0[0] = ok0 ? Br0[0]  : 0.f;  b0[1] = ok1 ? Br1[0]  : 0.f;
        b1[0] = ok0 ? Br0[16] : 0.f;  b1[1] = ok1 ? Br1[16] : 0.f;
        b2[0] = ok0 ? Br0[32] : 0.f;  b2[1] = ok1 ? Br1[32] : 0.f;
        b3[0] = ok0 ? Br0[48] : 0.f;  b3[1] = ok1 ? Br1[48] : 0.f;
        acc0 = __builtin_amdgcn_wmma_f32_16x16x4_f32(false, a, false, b0, (short)0, acc0, false, false);
        acc1 = __builtin_amdgcn_wmma_f32_16x16x4_f32(false, a, false, b1, (short)0, acc1, false, false);
        acc2 = __builtin_amdgcn_wmma_f32_16x16x4_f32(false, a, false, b2, (short)0, acc2, false, false);
        acc3 = __builtin_amdgcn_wmma_f32_16x16x4_f32(false, a, false, b3, (short)0, acc3, false, false);
    }

    const int nO = lane & 15;
    const int mO = (lane >> 4) << 3;
    float* Cb = C + (size_t)(m0 + mO) * N + n0 + nO;
    const v8f accs[4] = {acc0, acc1, acc2, acc3};
#pragma unroll
    for (int t = 0; t < 4; ++t) {
        const float bv = bias ? bias[n0 + t * 16 + nO] : 0.f;
#pragma unroll
        for (int r = 0; r < 8; ++r)
            Cb[(size_t)r * N + t * 16] = accs[t][r] + bv;
    }
}

// ---------------------------------------------------------------------------
// Softmax over the 16 (LVLS*PTS) logits per (bn, q, head), in place
// ---------------------------------------------------------------------------
__global__ void softmax16(float* __restrict__ aw) {
    const int i = blockIdx.x * blockDim.x + threadIdx.x;
    if (i >= BN * NQ * HEADS) return;
    float* p = aw + (size_t)i * 16;
    float mx = p[0];
#pragma unroll
    for (int j = 1; j < 16; ++j) mx = fmaxf(mx, p[j]);
    float s = 0.f;
#pragma unroll
    for (int j = 0; j < 16; ++j) { const float e = __expf(p[j] - mx); p[j] = e; s += e; }
    const float inv = 1.f / s;
#pragma unroll
    for (int j = 0; j < 16; ++j) p[j] *= inv;
}

// ---------------------------------------------------------------------------
// Deformable attention gather: block per (bn, q); 256 threads = (head, d)
// ---------------------------------------------------------------------------
__global__ void deform_attn(const float* __restrict__ val,
                            const float* __restrict__ rpc,
                            const float* __restrict__ off,
                            const float* __restrict__ aw,
                            float* __restrict__ attn) {
    const int bnq = blockIdx.x;                // 0 .. BN*NQ-1
    const int h = threadIdx.x >> 5;            // 0..7
    const int d = threadIdx.x & 31;            // 0..31
    const int bn = bnq / NQ;

    const float rx = rpc[bnq * 2 + 0];
    const float ry = rpc[bnq * 2 + 1];

    const int HWs[4][2]  = {{32, 112}, {16, 56}, {8, 28}, {4, 14}};
    const int starts[4]  = {0, 3584, 4480, 4704};

    float acc = 0.f;
#pragma unroll
    for (int l = 0; l < LVLS; ++l) {
        const int Hl = HWs[l][0], Wl = HWs[l][1], st = starts[l];
#pragma unroll
        for (int p = 0; p < PTS; ++p) {
            const int oidx = bnq * (HEADS * LVLS * PTS * 2) + (((h * LVLS + l) * PTS + p) * 2);
            const float ox = off[oidx], oy = off[oidx + 1];
            const float w  = aw[bnq * (HEADS * LVLS * PTS) + h * 16 + l * 4 + p];
            const float lx = rx + ox / (float)Wl;
            const float ly = ry + oy / (float)Hl;
            const float xx = lx * (float)Wl - 0.5f;
            const float yy = ly * (float)Hl - 0.5f;
            const int x0 = (int)floorf(xx), y0 = (int)floorf(yy);
            const float wx = xx - (float)x0, wy = yy - (float)y0;
            auto g = [&](int xi, int yi) -> float {
                if (xi < 0 || xi >= Wl || yi < 0 || yi >= Hl) return 0.f;
                return val[((size_t)bn * STOT + st + yi * Wl + xi) * DD + h * HD + d];
            };
            const float s = g(x0,   y0  ) * (1.f - wx) * (1.f - wy)
                          + g(x0+1, y0  ) * wx         * (1.f - wy)
                          + g(x0,   y0+1) * (1.f - wx) * wy
                          + g(x0+1, y0+1) * wx         * wy;
            acc += w * s;
        }
    }
    attn[(size_t)bnq * DD + h * HD + d] = acc;
}

// ---------------------------------------------------------------------------
// Finalize: mask, cross-camera mean (cnt), mean over NQ, broadcast over T,
// zero the lidar half.  thread per (b, d)
// ---------------------------------------------------------------------------
__global__ void finalize(const float* __restrict__ outp,
                         const float* __restrict__ maskf,
                         float* __restrict__ out) {
    const int i = blockIdx.x * blockDim.x + threadIdx.x;
    if (i >= BSZ * DD) return;
    const int b = i >> 8, d = i & 255;
    float acc = 0.f;
    for (int q = 0; q < NQ; ++q) {
        float cnt = 0.f, s = 0.f;
#pragma unroll
        for (int c = 0; c < NCAM; ++c) {
            const int row = (b * NCAM + c) * NQ + q;
            const float mk = maskf[row];
            cnt += mk;
            s += outp[(size_t)row * DD + d] * mk;
        }
        acc += s / fmaxf(cnt, 1.f);
    }
    acc *= (1.f / (float)NQ);
#pragma unroll
    for (int t = 0; t < TT; ++t) {
        out[((size_t)b * TT + t) * 512 + d]       = acc;
        out[((size_t)b * TT + t) * 512 + 256 + d] = 0.f;
    }
}

// ---------------------------------------------------------------------------
// Launcher
// ---------------------------------------------------------------------------
extern "C" void kernel_launch(void* const* d_in, const int* in_sizes, int n_in,
                              void* d_out, int out_size, void* d_ws, size_t ws_size,
                              hipStream_t stream) {
    const float* wp   = (const float*)d_in[0];
    const float* ctrl = (const float*)d_in[1];
    const float* meas = (const float*)d_in[2];
    const float* flat = (const float*)d_in[3];
    const float* l2i  = (const float*)d_in[4];
    const float* ida  = (const float*)d_in[5];
    const float* f0   = (const float*)d_in[6];
    const float* f1   = (const float*)d_in[7];
    const float* f2   = (const float*)d_in[8];
    const float* f3   = (const float*)d_in[9];
    const float* fpn  = (const float*)d_in[10];
    // d_in[11] lidar_feat: dead code (output lidar half is zeros)
    const float* temb = (const float*)d_in[12];
    const float* semb = (const float*)d_in[13];
    const int*   ih   = (const int*)d_in[14];
    const int*   iw   = (const int*)d_in[15];
    const float* Wq   = (const float*)d_in[16];
    const float* bq   = (const float*)d_in[17];
    const float* Wo   = (const float*)d_in[18];
    const float* bo   = (const float*)d_in[19];
    const float* Wa   = (const float*)d_in[20];
    const float* ba   = (const float*)d_in[21];
    const float* Wv   = (const float*)d_in[22];
    const float* bv   = (const float*)d_in[23];
    const float* Wout = (const float*)d_in[24];
    const float* bout = (const float*)d_in[25];
    // d_in[26..33]: lidar MLP weights, dead code

    float* ws = (float*)d_ws;
    const size_t ROWS = (size_t)BN * NQ;            // 4320
    float* qfull = ws;                              // ROWS * QDIM
    float* rpc   = qfull + ROWS * QDIM;             // ROWS * 2
    float* maskf = rpc   + ROWS * 2;                // ROWS
    float* qp    = maskf + ROWS;                    // ROWS * 256
    float* off   = qp    + ROWS * DD;               // ROWS * 256
    float* aw    = off   + ROWS * DD;               // ROWS * 128
    float* attn  = aw    + ROWS * 128;              // ROWS * 256
    float* outp  = attn  + ROWS * DD;               // ROWS * 256
    float* val   = outp  + ROWS * DD;               // BN*STOT*256 (156 MB)

    // 1) base query + projection
    build_query_proj<<<BSZ * NQ, 128, 0, stream>>>(wp, ctrl, meas, flat, l2i, ida,
                                                   temb, semb, ih, iw,
                                                   qfull, rpc, maskf);
    // 2) FPN bilinear sampling -> qfull tail
    {
        const int total = BN * NQ * LVLS * DD;
        sample_img_feats<<<(total + 255) / 256, 256, 0, stream>>>(f0, f1, f2, f3, rpc, qfull);
    }
    // 3) qp = qfull @ Wq + bq    (4320 x 1799 x 256)
    {
        dim3 grid(((BN * NQ) / 16 + 3) / 4, DD / 64);
        gemm_wmma_f32<<<grid, 128, 0, stream>>>(qfull, Wq, bq, qp, BN * NQ, DD, QDIM);
    }
    // 4) val = fpn @ Wv + bv     (152320 x 256 x 256)
    {
        dim3 grid(((BN * STOT) / 16 + 3) / 4, DD / 64);
        gemm_wmma_f32<<<grid, 128, 0, stream>>>(fpn, Wv, bv, val, BN * STOT, DD, DD);
    }
    // 5) off = qp @ Wo + bo      (4320 x 256 x 256)
    {
        dim3 grid(((BN * NQ) / 16 + 3) / 4, DD / 64);
        gemm_wmma_f32<<<grid, 128, 0, stream>>>(qp, Wo, bo, off, BN * NQ, DD, DD);
    }
    // 6) aw logits = qp @ Wa + ba (4320 x 256 x 128), then softmax over 16
    {
        dim3 grid(((BN * NQ) / 16 + 3) / 4, 128 / 64);
        gemm_wmma_f32<<<grid, 128, 0, stream>>>(qp, Wa, ba, aw, BN * NQ, 128, DD);
        const int n = BN * NQ * HEADS;
        softmax16<<<(n + 255) / 256, 256, 0, stream>>>(aw);
    }
    // 7) deformable gather
    deform_attn<<<BN * NQ, 256, 0, stream>>>(val, rpc, off, aw, attn);
    // 8) out = attn @ Wout + bout
    {
        dim3 grid(((BN * NQ) / 16 + 3) / 4, DD / 64);
        gemm_wmma_f32<<<grid, 128, 0, stream>>>(attn, Wout, bout, outp, BN * NQ, DD, DD);
    }
    // 9) finalize into d_out
    finalize<<<(BSZ * DD + 255) / 256, 256, 0, stream>>>(outp, maskf, (float*)d_out);
}